// OctreeTrivecRasterizer_50259707298380
// MI455X (gfx1250) — compile-verified
//
#include <hip/hip_runtime.h>
#include <math.h>

// ---------------- problem constants (from reference) ----------------
#define N_PTS   262144
#define R_CNT   8
#define D_CNT   8
#define SH_K    9          // (SH_DEG+1)^2 = 9
#define IMG_H   512
#define IMG_W   512
#define N_PIX   (IMG_H*IMG_W)
#define DENSITY_SHIFT (-2.0f)

// rows of the trivec tensor when viewed as (N*R*3) x D
#define MID_ROWS_TOTAL  ((unsigned long long)N_PTS * R_CNT * 3)   // 6,291,456
#define MID_ROWS_BLOCK  (256 * R_CNT * 3)                          // 6144 per block

typedef __attribute__((ext_vector_type(2))) float v2f;
typedef __attribute__((ext_vector_type(8))) float v8f;
typedef __attribute__((ext_vector_type(4))) unsigned int v4u;
typedef __attribute__((ext_vector_type(4))) int v4i;
typedef __attribute__((ext_vector_type(8))) int v8i;

// =====================================================================
// Kernel 1: per-point preprocess.
//  - TDM (tensor_load_to_lds): DMA the stride-8 trivec mid-slice column
//    for this block's 256 points (1x6144 tile, 24 KB) into LDS; the VALU
//    loop then reads dense LDS instead of 24 strided global loads/point.
//  - projection via V_WMMA_F32_16X16X4_F32:
//      A (16x4) = [wx wy wz 1] for 16 points, B (4x16) shared:
//      cols 0..3 = projmatrix columns, col 4 = view[:,2] (incl. row-3
//      translation since A[:,3] == 1) -> ph.xyzw and depth in one op.
//  - trivec density, alpha, SH deg-2 color (VALU, bandwidth bound)
//  - per-pixel histogram for the counting sort
// =====================================================================
__global__ __launch_bounds__(256)
void k_preprocess(const float* __restrict__ pos,
                  const float* __restrict__ trivecs,
                  const float* __restrict__ dens,
                  const float* __restrict__ shs,
                  const float* __restrict__ view,
                  const float* __restrict__ proj,
                  const float* __restrict__ campos,
                  const float* __restrict__ aabb,
                  int*   __restrict__ d_pid,
                  float* __restrict__ d_depth,
                  float* __restrict__ d_alpha,
                  float* __restrict__ d_c0,
                  float* __restrict__ d_c1,
                  float* __restrict__ d_c2,
                  int*   __restrict__ d_cnt)
{
    const int tid  = threadIdx.x;
    const int lane = tid & 31;
    const int wave = tid >> 5;
    const int n    = blockIdx.x * 256 + tid;        // N_PTS % 256 == 0

    __shared__ float smem[8][2][16][16];            // WMMA un-stripe, 16 KB
    __shared__ float ldsmid[MID_ROWS_BLOCK];        // TDM tile, 24 KB

    // ================= TDM: async tensor load of mid-slice =============
    // trivecs viewed as 2D tensor: MID_ROWS_TOTAL rows x 8 floats.
    // Tile: 1 element wide (column D/2), 6144 rows starting at this
    // block's first row -> packed dense into ldsmid[].
    if (wave == 0) {                                // uniform branch
        unsigned long long rowStart =
            (unsigned long long)blockIdx.x * MID_ROWS_BLOCK;
        unsigned long long ga = (unsigned long long)(uintptr_t)trivecs
                              + 4ull * (rowStart * D_CNT + (D_CNT / 2));
        unsigned int ldsOff = (unsigned int)(uintptr_t)&ldsmid[0];

        // ---- D# group 0 ----
        v4u g0;
        g0.x = 1u;                                   // count=1 (valid), user mode
        g0.y = ldsOff;                               // lds_addr
        g0.z = (unsigned int)(ga & 0xFFFFFFFFull);   // global_addr[31:0]
        g0.w = (unsigned int)((ga >> 32) & 0x1FFFFFFull)
             | (2u << 30);                           // global_addr[56:32] | type=2

        // ---- D# group 1 ----
        unsigned int td0 = D_CNT;                    // tensor_dim0 = 8 elements
        unsigned int td1 = (unsigned int)MID_ROWS_TOTAL;
        v8i g1;
        g1[0] = (int)(2u << 16);                     // data_size = 4 bytes
        g1[1] = (int)((td0 & 0xFFFFu) << 16);        // tensor_dim0[15:0]
        g1[2] = (int)(((td0 >> 16) & 0xFFFFu) | ((td1 & 0xFFFFu) << 16));
        g1[3] = (int)(((td1 >> 16) & 0xFFFFu) | (1u << 16));  // tile_dim0 = 1
        g1[4] = (int)(MID_ROWS_BLOCK & 0xFFFFu);     // tile_dim1 = 6144, tile_dim2=0
        g1[5] = (int)D_CNT;                          // tensor_dim0_stride = 8
        g1[6] = 0;                                   // stride hi / dim1_stride lo
        g1[7] = 0;

        v4i g2 = {};                                 // 2D tensor: groups 2-3 unused
        v4i g3 = {};
#if __clang_major__ >= 23
        v8i g4 = {};
        __builtin_amdgcn_tensor_load_to_lds(g0, g1, g2, g3, g4, 0);
#else
        __builtin_amdgcn_tensor_load_to_lds(g0, g1, g2, g3, 0);
#endif
        __builtin_amdgcn_s_wait_tensorcnt(0);        // s_wait_tensorcnt 0
    }
    __syncthreads();                                 // publish LDS tile

    // ---- world position (branch-free: EXEC stays all-1 for WMMA) ----
    float px = pos[n*3+0], py = pos[n*3+1], pz = pos[n*3+2];
    float wx = aabb[0] + px * aabb[3];
    float wy = aabb[1] + py * aabb[4];
    float wz = aabb[2] + pz * aabb[5];

    // prefetch the SH block for this point (global_prefetch_b8)
    __builtin_prefetch(&shs[(size_t)n * R_CNT * SH_K * 3], 0, 1);

    // ---- build WMMA operands -----------------------------------------
    // A layout (32-bit, 16x4): VGPR0 = K0 (lanes 0-15) / K2 (lanes 16-31)
    //                          VGPR1 = K1              / K3
    const int  nlo = lane & 15;
    const bool hi  = lane >= 16;
    float s_wz0 = __shfl(wz, nlo);        // z of points 0..15  (for lanes>=16, tile0)
    float s_wx1 = __shfl(wx, nlo + 16);   // x of points 16..31 (for lanes<16,  tile1)
    float s_wy1 = __shfl(wy, nlo + 16);

    v2f A0, A1, B;
    A0.x = hi ? s_wz0 : wx;    A0.y = hi ? 1.0f : wy;     // tile0: points 0..15
    A1.x = hi ? wz    : s_wx1; A1.y = hi ? 1.0f : s_wy1;  // tile1: points 16..31

    // B layout (4x16): VGPR0 = row K0 (lanes 0-15) / K2 (lanes 16-31), VGPR1 = K1/K3
    int   k0 = hi ? 2 : 0;
    float b0 = 0.0f, b1 = 0.0f;
    if (nlo < 4)       { b0 = proj[k0*4 + nlo]; b1 = proj[(k0+1)*4 + nlo]; }
    else if (nlo == 4) { b0 = view[k0*4 + 2];   b1 = view[(k0+1)*4 + 2];   }
    B.x = b0; B.y = b1;

    v8f C0 = {}; v8f C1 = {};
    C0 = __builtin_amdgcn_wmma_f32_16x16x4_f32(false, A0, false, B, (short)0, C0, false, false);
    C1 = __builtin_amdgcn_wmma_f32_16x16x4_f32(false, A1, false, B, (short)0, C1, false, false);

    // ---- un-stripe C through LDS: lane p needs row p, cols 0..4 ------
    const int rbase = hi ? 8 : 0;
#pragma unroll
    for (int v = 0; v < 8; ++v) {
        smem[wave][0][v + rbase][nlo] = C0[v];
        smem[wave][1][v + rbase][nlo] = C1[v];
    }
    __syncthreads();
    const float* rowp = &smem[wave][hi ? 1 : 0][nlo][0];
    float phx = rowp[0], phy = rowp[1], phw = rowp[3];
    float depth = rowp[4];

    // ---- screen position / pixel id ----------------------------------
    float invw = 1.0f / (phw + 1e-7f);
    float ndx = phx * invw, ndy = phy * invw;
    int ix = (int)rintf(((ndx + 1.0f) * (float)IMG_W - 1.0f) * 0.5f);
    int iy = (int)rintf(((ndy + 1.0f) * (float)IMG_H - 1.0f) * 0.5f);
    bool valid = (depth > 0.2f) && (ix >= 0) && (ix < IMG_W) && (iy >= 0) && (iy < IMG_H);
    int pid = valid ? (iy * IMG_W + ix) : N_PIX;

    // ---- trivec density (mids from the TDM-staged LDS tile) ----------
    const float* mids = &ldsmid[tid * (R_CNT * 3)];
    const float* dn   = dens + (size_t)n * R_CNT;
    const float* sh   = shs  + (size_t)n * R_CNT * SH_K * 3;

    float sr[R_CNT];
    float sigma = 0.0f;
#pragma unroll
    for (int r = 0; r < R_CNT; ++r) {
        float m0 = mids[r*3 + 0];
        float m1 = mids[r*3 + 1];
        float m2 = mids[r*3 + 2];
        float tri = fmaxf(m0 * m1 * m2, 0.0f);
        float s = expf(dn[r] + DENSITY_SHIFT) * tri;
        sr[r] = s;
        sigma += s;
    }
    float alpha = 1.0f - expf(-sigma);
    alpha = fminf(fmaxf(alpha, 0.0f), 0.999f);

    // ---- SH deg-2 color ----------------------------------------------
    float dx = wx - campos[0], dy = wy - campos[1], dz = wz - campos[2];
    float invn = 1.0f / (sqrtf(dx*dx + dy*dy + dz*dz) + 1e-8f);
    float x = dx * invn, y = dy * invn, z = dz * invn;
    float basis[SH_K];
    basis[0] =  0.28209479177387814f;
    basis[1] = -0.4886025119029199f * y;
    basis[2] =  0.4886025119029199f * z;
    basis[3] = -0.4886025119029199f * x;
    basis[4] =  1.0925484305920792f * x * y;
    basis[5] = -1.0925484305920792f * y * z;
    basis[6] =  0.31539156525252005f * (2.0f*z*z - x*x - y*y);
    basis[7] = -1.0925484305920792f * x * z;
    basis[8] =  0.5462742152960396f * (x*x - y*y);

    float invsig = 1.0f / (sigma + 1e-8f);
    float c0 = 0.0f, c1 = 0.0f, c2 = 0.0f;
#pragma unroll
    for (int r = 0; r < R_CNT; ++r) {
        float a0 = 0.0f, a1 = 0.0f, a2 = 0.0f;
#pragma unroll
        for (int k = 0; k < SH_K; ++k) {
            const float* sp = sh + ((size_t)r*SH_K + k) * 3;
            a0 += basis[k] * sp[0];
            a1 += basis[k] * sp[1];
            a2 += basis[k] * sp[2];
        }
        a0 = fmaxf(a0 + 0.5f, 0.0f);
        a1 = fmaxf(a1 + 0.5f, 0.0f);
        a2 = fmaxf(a2 + 0.5f, 0.0f);
        float wr = sr[r] * invsig;
        c0 += wr * a0; c1 += wr * a1; c2 += wr * a2;
    }

    d_pid[n]   = pid;
    d_depth[n] = depth;
    d_alpha[n] = alpha;
    d_c0[n] = c0; d_c1[n] = c1; d_c2[n] = c2;
    atomicAdd(&d_cnt[pid], 1);
}

// =====================================================================
// Counting-sort scan machinery (exclusive scan of N_PIX+1 counters)
// =====================================================================
__global__ __launch_bounds__(256)
void k_scan_blocks(const int* __restrict__ cnt, int* __restrict__ partial,
                   int* __restrict__ bsum, int ntot)
{
    __shared__ int sm[256];
    int gid = blockIdx.x * 256 + threadIdx.x;
    int v = (gid < ntot) ? cnt[gid] : 0;
    sm[threadIdx.x] = v;
    __syncthreads();
#pragma unroll
    for (int off = 1; off < 256; off <<= 1) {
        int t = (threadIdx.x >= off) ? sm[threadIdx.x - off] : 0;
        __syncthreads();
        sm[threadIdx.x] += t;
        __syncthreads();
    }
    int incl = sm[threadIdx.x];
    if (gid < ntot) partial[gid] = incl - v;        // exclusive within block
    if (threadIdx.x == 255) bsum[blockIdx.x] = incl;
}

__global__ void k_scan_sums(int* __restrict__ bsum, int nb)
{
    if (blockIdx.x == 0 && threadIdx.x == 0) {
        int acc = 0;
        for (int i = 0; i < nb; ++i) { int t = bsum[i]; bsum[i] = acc; acc += t; }
    }
}

__global__ __launch_bounds__(256)
void k_add_sums(const int* __restrict__ partial, const int* __restrict__ bsum,
                int* __restrict__ start, int* __restrict__ cursor, int ntot)
{
    int gid = blockIdx.x * 256 + threadIdx.x;
    if (gid < ntot) {
        int v = partial[gid] + bsum[blockIdx.x];
        start[gid]  = v;
        cursor[gid] = v;
    }
}

__global__ __launch_bounds__(256)
void k_scatter(const int* __restrict__ pid, int* __restrict__ cursor,
               int* __restrict__ ord, int npts)
{
    int gid = blockIdx.x * 256 + threadIdx.x;
    if (gid < npts) {
        int dst = atomicAdd(&cursor[pid[gid]], 1);
        ord[dst] = gid;
    }
}

// =====================================================================
// Kernel: per-pixel depth-ordered alpha compositing.
// Enumerates the pixel's segment in lexicographic (depth, slot) order
// (selection traversal, O(k^2); avg segment length ~1 here).
// =====================================================================
__global__ __launch_bounds__(256)
void k_composite(const int* __restrict__ start, const int* __restrict__ ord,
                 const float* __restrict__ depth, const float* __restrict__ alpha,
                 const float* __restrict__ c0a, const float* __restrict__ c1a,
                 const float* __restrict__ c2a, const float* __restrict__ bg,
                 float* __restrict__ out)
{
    int p = blockIdx.x * 256 + threadIdx.x;
    if (p >= N_PIX) return;
    int s = start[p], e = start[p + 1];

    float T = 1.0f, C0 = 0.0f, C1 = 0.0f, C2 = 0.0f, A = 0.0f, Dw = 0.0f;
    float lastD = -3.402823e38f;
    int lastJ = -1;
    for (int it = 0; it < (e - s); ++it) {
        float bestD = 3.402823e38f;
        int bestJ = 2147483647, bestN = -1;
        for (int j = s; j < e; ++j) {
            int nn = ord[j];
            float dj = depth[nn];
            bool after  = (dj > lastD) || (dj == lastD && j > lastJ);
            bool better = (dj < bestD) || (dj == bestD && j < bestJ);
            if (after && better) { bestD = dj; bestJ = j; bestN = nn; }
        }
        if (bestN < 0) break;
        float al = alpha[bestN];
        float w  = al * T;
        C0 += w * c0a[bestN];
        C1 += w * c1a[bestN];
        C2 += w * c2a[bestN];
        A  += w;
        Dw += w * bestD;
        T *= (1.0f - al);
        lastD = bestD; lastJ = bestJ;
    }
    float b0 = bg[0], b1 = bg[1], b2 = bg[2];
    out[0 * N_PIX + p] = C0 + (1.0f - A) * b0;   // img_c channel 0
    out[1 * N_PIX + p] = C1 + (1.0f - A) * b1;   // img_c channel 1
    out[2 * N_PIX + p] = C2 + (1.0f - A) * b2;   // img_c channel 2
    out[3 * N_PIX + p] = Dw;                     // img_d
    out[4 * N_PIX + p] = A;                      // img_a
    out[5 * N_PIX + p] = Dw / (A + 1e-8f);       // pct_d
}

// =====================================================================
extern "C" void kernel_launch(void* const* d_in, const int* in_sizes, int n_in,
                              void* d_out, int out_size, void* d_ws, size_t ws_size,
                              hipStream_t stream)
{
    (void)in_sizes; (void)n_in; (void)out_size; (void)ws_size;

    const float* pos     = (const float*)d_in[0];
    const float* trivecs = (const float*)d_in[1];
    const float* dens    = (const float*)d_in[2];
    const float* shs     = (const float*)d_in[3];
    const float* view    = (const float*)d_in[4];
    const float* proj    = (const float*)d_in[5];
    const float* campos  = (const float*)d_in[6];
    const float* aabb    = (const float*)d_in[7];
    const float* bg      = (const float*)d_in[8];
    // d_in[9] = random_image (unused by reference math)
    float* out = (float*)d_out;

    // ---- workspace layout (~11.5 MB) ----
    char* w = (char*)d_ws;
    int*   d_pid     = (int*)w;   w += (size_t)N_PTS * 4;
    float* d_depth   = (float*)w; w += (size_t)N_PTS * 4;
    float* d_alpha   = (float*)w; w += (size_t)N_PTS * 4;
    float* d_c0      = (float*)w; w += (size_t)N_PTS * 4;
    float* d_c1      = (float*)w; w += (size_t)N_PTS * 4;
    float* d_c2      = (float*)w; w += (size_t)N_PTS * 4;
    int*   d_cnt     = (int*)w;   w += (size_t)(N_PIX + 1) * 4;
    int*   d_partial = (int*)w;   w += (size_t)(N_PIX + 1) * 4;
    int*   d_bsum    = (int*)w;   w += (size_t)2048 * 4;
    int*   d_start   = (int*)w;   w += (size_t)(N_PIX + 1) * 4;
    int*   d_cursor  = (int*)w;   w += (size_t)(N_PIX + 1) * 4;
    int*   d_ord     = (int*)w;   w += (size_t)N_PTS * 4;

    hipMemsetAsync(d_cnt, 0, (size_t)(N_PIX + 1) * sizeof(int), stream);

    k_preprocess<<<N_PTS / 256, 256, 0, stream>>>(
        pos, trivecs, dens, shs, view, proj, campos, aabb,
        d_pid, d_depth, d_alpha, d_c0, d_c1, d_c2, d_cnt);

    const int nScan = N_PIX + 1;
    const int nBlk  = (nScan + 255) / 256;    // 1025
    k_scan_blocks<<<nBlk, 256, 0, stream>>>(d_cnt, d_partial, d_bsum, nScan);
    k_scan_sums<<<1, 1, 0, stream>>>(d_bsum, nBlk);
    k_add_sums<<<nBlk, 256, 0, stream>>>(d_partial, d_bsum, d_start, d_cursor, nScan);
    k_scatter<<<N_PTS / 256, 256, 0, stream>>>(d_pid, d_cursor, d_ord, N_PTS);
    k_composite<<<N_PIX / 256, 256, 0, stream>>>(
        d_start, d_ord, d_depth, d_alpha, d_c0, d_c1, d_c2, bg, out);
}